// Graph_Generator_12206297055752
// MI455X (gfx1250) — compile-verified
//
#include <hip/hip_runtime.h>

typedef __attribute__((ext_vector_type(2))) float v2f;
typedef __attribute__((ext_vector_type(8))) float v8f;

#define NN 2048
#define CC 128
#define TT 288
#define KTOP 1638            // int(2048 * 0.8)
#define INV_SCALE 0.022097086912079608f   // 1/sqrt(2048)

// ---------------------------------------------------------------------------
// Kernel 1: x_sum[c,n] = sum_t x[c,n,t].  One wave per (c,n) pair, lanes
// stride the contiguous T axis (coalesced 128B bursts), then wave32 shfl tree.
// ---------------------------------------------------------------------------
__global__ void reduceT_kernel(const float* __restrict__ x, float* __restrict__ xs) {
    const int lane = threadIdx.x & 31;
    const int wave = threadIdx.x >> 5;
    const long long p = (long long)blockIdx.x * 8 + wave;   // p = c*NN + n
    const long long base = p * TT;
    float s = 0.0f;
    #pragma unroll
    for (int t = lane; t < TT; t += 32) s += x[base + t];
    #pragma unroll
    for (int o = 16; o > 0; o >>= 1) s += __shfl_xor(s, o, 32);
    if (lane == 0) xs[p] = s;
}

// ---------------------------------------------------------------------------
// Kernel 2: dual GEMM via V_WMMA_F32_16X16X4_F32.
//   s1[n,m] = relu( (sum_c xs[c,n]*mem[c,m]) / sqrt(N) )
//   s2[n,m] = relu( (sum_c xs[c,n]*xs [c,m]) / sqrt(N) )
// Workgroup = 8 waves, each wave owns one 16x16 tile; workgroup tile 16x128.
// A-fragment (xs rows) is shared by both WMMAs each K-step.
//
// f32 16x4 A layout (ISA 7.12.2): lanes 0-15 -> M=0..15 {K=0 in v0, K=1 in v1},
// lanes 16-31 -> same M, K=2/3.  B (4x16) mirrors it with N striped across
// lanes, so both fragments use the same addressing into a [c][n] array.
// ---------------------------------------------------------------------------
__global__ void gemm_wmma_kernel(const float* __restrict__ xs,
                                 const float* __restrict__ mem,
                                 float* __restrict__ s1,
                                 float* __restrict__ s2) {
    const int lane = threadIdx.x & 31;
    const int wave = threadIdx.x >> 5;
    const int half = lane >> 4;      // 0: K=0,1   1: K=2,3
    const int l15  = lane & 15;
    const int n0 = blockIdx.y * 16;              // output row block
    const int m0 = blockIdx.x * 128 + wave * 16; // output col block (per wave)

    v8f acc1 = {};
    v8f acc2 = {};

    for (int c0 = 0; c0 < CC; c0 += 4) {
        const int rA = (c0 + 2 * half) * NN;
        const int rB = rA + NN;
        v2f a, b1, b2;
        a.x  = xs [rA + n0 + l15];  a.y  = xs [rB + n0 + l15];
        b1.x = mem[rA + m0 + l15];  b1.y = mem[rB + m0 + l15];
        b2.x = xs [rA + m0 + l15];  b2.y = xs [rB + m0 + l15];
        acc1 = __builtin_amdgcn_wmma_f32_16x16x4_f32(
            false, a, false, b1, (short)0, acc1, false, false);
        acc2 = __builtin_amdgcn_wmma_f32_16x16x4_f32(
            false, a, false, b2, (short)0, acc2, false, false);
    }

    // C/D layout: VGPR r holds row M = r + 8*half, N = lane&15.
    #pragma unroll
    for (int r = 0; r < 8; ++r) {
        const int row = n0 + r + 8 * half;
        const int col = m0 + l15;
        s1[(long long)row * NN + col] = fmaxf(acc1[r] * INV_SCALE, 0.0f);
        s2[(long long)row * NN + col] = fmaxf(acc2[r] * INV_SCALE, 0.0f);
    }
}

// ---------------------------------------------------------------------------
// Block reduction helpers (256 threads = 8 waves, wave32).
// ---------------------------------------------------------------------------
__device__ __forceinline__ float blkMax(float v, float* lds) {
    #pragma unroll
    for (int o = 16; o > 0; o >>= 1) v = fmaxf(v, __shfl_xor(v, o, 32));
    __syncthreads();
    if ((threadIdx.x & 31) == 0) lds[threadIdx.x >> 5] = v;
    __syncthreads();
    float r = lds[0];
    #pragma unroll
    for (int i = 1; i < 8; ++i) r = fmaxf(r, lds[i]);
    return r;
}

__device__ __forceinline__ float blkSumF(float v, float* lds) {
    #pragma unroll
    for (int o = 16; o > 0; o >>= 1) v += __shfl_xor(v, o, 32);
    __syncthreads();
    if ((threadIdx.x & 31) == 0) lds[threadIdx.x >> 5] = v;
    __syncthreads();
    float r = lds[0];
    #pragma unroll
    for (int i = 1; i < 8; ++i) r += lds[i];
    return r;
}

__device__ __forceinline__ unsigned blkSumU(unsigned v, unsigned* lds) {
    #pragma unroll
    for (int o = 16; o > 0; o >>= 1) v += __shfl_xor(v, o, 32);
    __syncthreads();
    if ((threadIdx.x & 31) == 0) lds[threadIdx.x >> 5] = v;
    __syncthreads();
    unsigned r = lds[0];
    #pragma unroll
    for (int i = 1; i < 8; ++i) r += lds[i];
    return r;
}

// ---------------------------------------------------------------------------
// Kernel 3: one block per row.  Fused: softmax(s1), softmax(s2), linear
// combine, softmax again, then top-k(1638) mask with lowest-index
// tie-breaking (radix binary search on positive-float bit patterns).
// Each thread owns 8 *contiguous* columns so index-ordered tie ranking is a
// block prefix scan over thread chunks.
// ---------------------------------------------------------------------------
__global__ void softmax_topk_kernel(const float* __restrict__ s1,
                                    const float* __restrict__ s2,
                                    const float* __restrict__ fc_w,
                                    const float* __restrict__ fc_b,
                                    float* __restrict__ out) {
    __shared__ float    fred[8];
    __shared__ unsigned ured[8];
    __shared__ unsigned scan[256];

    const int row = blockIdx.x;
    const int tid = threadIdx.x;
    const float w0 = fc_w[0], w1 = fc_w[1], bb = fc_b[0];

    const long long base = (long long)row * NN + tid * 8;
    float r1[8], r2[8];
    #pragma unroll
    for (int j = 0; j < 8; ++j) { r1[j] = s1[base + j]; r2[j] = s2[base + j]; }

    // softmax of s1 row
    float lm = r1[0];
    #pragma unroll
    for (int j = 1; j < 8; ++j) lm = fmaxf(lm, r1[j]);
    float m1 = blkMax(lm, fred);
    float ls = 0.0f;
    #pragma unroll
    for (int j = 0; j < 8; ++j) { r1[j] = __expf(r1[j] - m1); ls += r1[j]; }
    float inv1 = 1.0f / blkSumF(ls, fred);

    // softmax of s2 row
    lm = r2[0];
    #pragma unroll
    for (int j = 1; j < 8; ++j) lm = fmaxf(lm, r2[j]);
    float m2 = blkMax(lm, fred);
    ls = 0.0f;
    #pragma unroll
    for (int j = 0; j < 8; ++j) { r2[j] = __expf(r2[j] - m2); ls += r2[j]; }
    float inv2 = 1.0f / blkSumF(ls, fred);

    // combine + final softmax
    float f[8];
    #pragma unroll
    for (int j = 0; j < 8; ++j) f[j] = r1[j] * inv1 * w0 + r2[j] * inv2 * w1 + bb;
    lm = f[0];
    #pragma unroll
    for (int j = 1; j < 8; ++j) lm = fmaxf(lm, f[j]);
    float mf = blkMax(lm, fred);
    ls = 0.0f;
    #pragma unroll
    for (int j = 0; j < 8; ++j) { f[j] = __expf(f[j] - mf); ls += f[j]; }
    float invf = 1.0f / blkSumF(ls, fred);
    float adj[8];
    unsigned bits[8];
    #pragma unroll
    for (int j = 0; j < 8; ++j) {
        adj[j]  = f[j] * invf;                 // strictly > 0
        bits[j] = __float_as_uint(adj[j]);     // order-preserving for +floats
    }

    // ---- k-th largest via binary search on bit patterns ----
    // V = minimal t with count(bits > t) <= KTOP-1  ==> V is the k-th largest.
    unsigned lo = 0u, hi = 0x7f800000u;
    while (lo < hi) {
        const unsigned mid = lo + ((hi - lo) >> 1);
        unsigned c = 0;
        #pragma unroll
        for (int j = 0; j < 8; ++j) c += (bits[j] > mid) ? 1u : 0u;
        c = blkSumU(c, ured);
        if (c <= (unsigned)(KTOP - 1)) hi = mid; else lo = mid + 1;
    }
    const unsigned V = lo;

    unsigned cgt = 0;
    #pragma unroll
    for (int j = 0; j < 8; ++j) cgt += (bits[j] > V) ? 1u : 0u;
    cgt = blkSumU(cgt, ured);
    const unsigned need = (unsigned)KTOP - cgt;   // ties to keep, by index

    // exclusive prefix of per-thread equal-counts (index order = thread order)
    unsigned eq = 0;
    #pragma unroll
    for (int j = 0; j < 8; ++j) eq += (bits[j] == V) ? 1u : 0u;
    __syncthreads();
    scan[tid] = eq;
    __syncthreads();
    for (int off = 1; off < 256; off <<= 1) {
        unsigned t = (tid >= off) ? scan[tid - off] : 0u;
        __syncthreads();
        scan[tid] += t;
        __syncthreads();
    }
    unsigned rank = scan[tid] - eq;   // equals with smaller index

    #pragma unroll
    for (int j = 0; j < 8; ++j) {
        bool keep = (bits[j] > V);
        if (bits[j] == V) { keep = (rank < need); ++rank; }
        out[base + j] = keep ? adj[j] : 0.0f;
    }
}

// ---------------------------------------------------------------------------
extern "C" void kernel_launch(void* const* d_in, const int* in_sizes, int n_in,
                              void* d_out, int out_size, void* d_ws, size_t ws_size,
                              hipStream_t stream) {
    const float* x    = (const float*)d_in[0];   // (128, 2048, 288)
    const float* mem  = (const float*)d_in[1];   // (128, 2048)
    const float* fc_w = (const float*)d_in[2];   // (1, 2)
    const float* fc_b = (const float*)d_in[3];   // (1,)
    float* out = (float*)d_out;                  // (2048, 2048)

    float* xs = (float*)d_ws;                    // (128, 2048)      1 MB
    float* s1 = xs + (size_t)CC * NN;            // (2048, 2048)    16 MB
    float* s2 = s1 + (size_t)NN * NN;            // (2048, 2048)    16 MB

    // 1) T-reduction: 262144 (c,n) pairs, 8 waves/block
    reduceT_kernel<<<(CC * NN) / 8, 256, 0, stream>>>(x, xs);

    // 2) dual f32 WMMA GEMM: grid (cols/128, rows/16)
    dim3 g2(NN / 128, NN / 16);
    gemm_wmma_kernel<<<g2, 256, 0, stream>>>(xs, mem, s1, s2);

    // 3) fused softmax chain + top-k mask, one block per row
    softmax_topk_kernel<<<NN, 256, 0, stream>>>(s1, s2, fc_w, fc_b, out);
}